// GateDSV2_42322607735337
// MI455X (gfx1250) — compile-verified
//
#include <hip/hip_runtime.h>
#include <math.h>

typedef __attribute__((ext_vector_type(2))) float v2f;
typedef __attribute__((ext_vector_type(4))) float v4f;
typedef __attribute__((ext_vector_type(8))) float v8f;

#define BN        32768
#define DIM       2048
#define NEXP      64
#define TOPK      6
#define NGROUPS   8
#define TOPKG     4
#define ROWS_PER_WAVE  32
#define NWAVES         4
#define ROWS_PER_BLOCK (ROWS_PER_WAVE * NWAVES)   // 128

// Fused DeepSeek-V2 router:
//   fp32 WMMA GEMM (x @ W^T, one wave = 32 rows x 64 experts, b128-fed)
//   -> LDS -> per-row softmax + group-limited top-k -> coalesced outputs.
__global__ __launch_bounds__(128) void GateDSV2_kernel(
    const float* __restrict__ x,     // [BN, DIM]
    const float* __restrict__ W,     // [NEXP, DIM]
    float* __restrict__ out_w,       // [BN, TOPK]
    int*   __restrict__ out_idx,     // [BN, TOPK]
    float* __restrict__ out_p)       // [BN, NEXP]
{
    __shared__ float lgt[ROWS_PER_BLOCK * NEXP];  // 32 KB logits tile
    __shared__ float row_max[ROWS_PER_BLOCK];
    __shared__ float row_inv[ROWS_PER_BLOCK];

    const int tid  = threadIdx.x;
    const int wv   = tid >> 5;
    const int lane = tid & 31;
    const int n    = lane & 15;      // A: row-in-16-tile, B: expert-in-16-tile
    const int h    = lane >> 4;      // half-wave: selects K quad {4h..4h+3}
    const int row0 = blockIdx.x * ROWS_PER_BLOCK;
    const int wrow = row0 + wv * ROWS_PER_WAVE;

    // K-permuted fragments: one b128 per lane covers K=8 for a WMMA pair.
    // WMMA-lo consumes logical K {k,k+1,k+4,k+5}, WMMA-hi {k+2,k+3,k+6,k+7};
    // the same permutation is applied to A and B, so the K-sum is unchanged.
    const float* aptr0 = x + (size_t)(wrow + n) * DIM + 4 * h;        // rows s=0
    const float* aptr1 = aptr0 + (size_t)16 * DIM;                    // rows s=1
    const float* bptr  = W + (size_t)n * DIM + 4 * h;                 // expert tile t at +t*16*DIM

    v8f acc[4][2];
#pragma unroll
    for (int t = 0; t < 4; ++t)
#pragma unroll
        for (int s = 0; s < 2; ++s)
            acc[t][s] = (v8f){0.f, 0.f, 0.f, 0.f, 0.f, 0.f, 0.f, 0.f};

    for (int k = 0; k < DIM; k += 8) {
        __builtin_prefetch(aptr0 + k + 32, 0, 1);    // gfx1250 global_prefetch_b8
        v4f a0 = *(const v4f*)(aptr0 + k);
        v4f a1 = *(const v4f*)(aptr1 + k);
        v2f a0lo = __builtin_shufflevector(a0, a0, 0, 1);
        v2f a0hi = __builtin_shufflevector(a0, a0, 2, 3);
        v2f a1lo = __builtin_shufflevector(a1, a1, 0, 1);
        v2f a1hi = __builtin_shufflevector(a1, a1, 2, 3);
#pragma unroll
        for (int t = 0; t < 4; ++t) {
            v4f b = *(const v4f*)(bptr + (size_t)(t * 16) * DIM + k);
            v2f blo = __builtin_shufflevector(b, b, 0, 1);
            v2f bhi = __builtin_shufflevector(b, b, 2, 3);
            acc[t][0] = __builtin_amdgcn_wmma_f32_16x16x4_f32(false, a0lo, false, blo,
                            (short)0, acc[t][0], false, false);
            acc[t][0] = __builtin_amdgcn_wmma_f32_16x16x4_f32(false, a0hi, false, bhi,
                            (short)0, acc[t][0], false, false);
            acc[t][1] = __builtin_amdgcn_wmma_f32_16x16x4_f32(false, a1lo, false, blo,
                            (short)0, acc[t][1], false, false);
            acc[t][1] = __builtin_amdgcn_wmma_f32_16x16x4_f32(false, a1hi, false, bhi,
                            (short)0, acc[t][1], false, false);
        }
    }

    // C/D layout: VGPR r -> M=r (lanes 0-15) / M=r+8 (lanes 16-31); N = lane&15
#pragma unroll
    for (int t = 0; t < 4; ++t)
#pragma unroll
        for (int s = 0; s < 2; ++s)
#pragma unroll
            for (int r = 0; r < 8; ++r) {
                int mm = s * 16 + r + 8 * h;
                lgt[(wv * ROWS_PER_WAVE + mm) * NEXP + t * 16 + n] = acc[t][s][r];
            }
    __syncthreads();

    // ---- per-row routing: one row per thread (all 128 threads active) ----
    {
        const float* l = lgt + tid * NEXP;

        // fp32 softmax stats over all 64 experts (original activations)
        float maxv = -INFINITY;
        for (int i = 0; i < NEXP; ++i) maxv = fmaxf(maxv, l[i]);
        float sum = 0.f;
        for (int i = 0; i < NEXP; ++i) sum += __expf(l[i] - maxv);
        const float inv = 1.0f / sum;
        row_max[tid] = maxv;
        row_inv[tid] = inv;

        // group scores = max logit per group of 8
        float gs[NGROUPS];
#pragma unroll
        for (int g = 0; g < NGROUPS; ++g) {
            float gm = l[g * 8];
#pragma unroll
            for (int j = 1; j < 8; ++j) gm = fmaxf(gm, l[g * 8 + j]);
            gs[g] = gm;
        }

        // top-4 groups (strict > keeps lowest index on ties, matching lax.top_k)
        unsigned keepg = 0u;
        for (int t = 0; t < TOPKG; ++t) {
            int bi = 0; float bv = -INFINITY;
#pragma unroll
            for (int g = 0; g < NGROUPS; ++g)
                if (((keepg >> g) & 1u) == 0u && gs[g] > bv) { bv = gs[g]; bi = g; }
            keepg |= (1u << bi);
        }

        // top-6 experts over group-masked logits (softmax is monotone -> same order)
        const int row = row0 + tid;
        unsigned long long taken = 0ull;
        for (int t = 0; t < TOPK; ++t) {
            int bi = 0; float bv = -INFINITY;
            for (int i = 0; i < NEXP; ++i) {
                if ((taken >> i) & 1ull) continue;
                float v = ((keepg >> (i >> 3)) & 1u) ? l[i] : -INFINITY;
                if (v > bv) { bv = v; bi = i; }
            }
            taken |= (1ull << bi);
            out_idx[(size_t)row * TOPK + t] = bi;
            // gather from ORIGINAL softmax; ROUTE_SCALE == 1.0
            out_w[(size_t)row * TOPK + t] = __expf(l[bi] - maxv) * inv;
        }
    }
    __syncthreads();

    // ---- coalesced probs store: idx = j*128 + tid is contiguous in memory ----
    {
        float* po = out_p + (size_t)row0 * NEXP;
        for (int j = 0; j < 64; ++j) {
            int idx = j * ROWS_PER_BLOCK + tid;
            int r   = idx >> 6;    // row within block
            po[idx] = __expf(lgt[idx] - row_max[r]) * row_inv[r];
        }
    }
}

extern "C" void kernel_launch(void* const* d_in, const int* in_sizes, int n_in,
                              void* d_out, int out_size, void* d_ws, size_t ws_size,
                              hipStream_t stream) {
    (void)in_sizes; (void)n_in; (void)out_size; (void)d_ws; (void)ws_size;
    const float* x = (const float*)d_in[0];
    const float* W = (const float*)d_in[1];

    // d_out: weights [BN*TOPK] f32 | indices [BN*TOPK] i32 | probs [BN*NEXP] f32
    float* out   = (float*)d_out;
    float* out_w = out;
    int*   out_i = (int*)(out + (size_t)BN * TOPK);
    float* out_p = out + (size_t)BN * TOPK * 2;

    dim3 grid(BN / ROWS_PER_BLOCK);   // 256 blocks
    dim3 block(ROWS_PER_BLOCK);       // 128 threads = 4 waves
    GateDSV2_kernel<<<grid, block, 0, stream>>>(x, W, out_w, out_i, out_p);
}